// E3_TransformerLayer_multi_66881230733950
// MI455X (gfx1250) — compile-verified
//
#include <hip/hip_runtime.h>

typedef __attribute__((ext_vector_type(2))) float v2f;
typedef __attribute__((ext_vector_type(8))) float v8f;

#define N_NODES 20000
#define N_EDGES 160000
#define OUT_CH  36
#define WPB     4           // waves (16-edge tiles) per block

__device__ __forceinline__ void lds_fence() {
  asm volatile("s_wait_dscnt 0" ::: "memory");
}

__device__ __forceinline__ float silu_f(float x) {
  return x / (1.0f + __expf(-x));
}

// ---------------------------------------------------------------- zero
__global__ void e3_zero_kernel(float* __restrict__ out, float* __restrict__ cnt) {
  int i = blockIdx.x * blockDim.x + threadIdx.x;
  if (i < N_NODES * OUT_CH) out[i] = 0.0f;
  if (i < N_NODES) cnt[i] = 0.0f;
}

// ------------------------------------------------- pack weights to WMMA B-frag order
// Wp[((ks*NT + nt)*32 + lane)*2 + j] = W[(4*ks + 2*(lane>>4) + j)*N + nt*16 + (lane&15)]
__global__ void e3_pack_kernel(const float* __restrict__ W, float* __restrict__ Wp,
                               int K, int N) {
  int i = blockIdx.x * blockDim.x + threadIdx.x;
  if (i >= K * N) return;
  int j    = i & 1;
  int lane = (i >> 1) & 31;
  int grp  = i >> 6;
  int NT   = N >> 4;
  int nt   = grp % NT;
  int ks   = grp / NT;
  int k    = 4 * ks + 2 * (lane >> 4) + j;
  int col  = nt * 16 + (lane & 15);
  Wp[i] = W[k * N + col];
}

// ---------------------------------------------------------------- node MLP -> Ai
__global__ void e3_node_mlp_kernel(const float* __restrict__ emb,
                                   const float* __restrict__ w1, const float* __restrict__ b1,
                                   const float* __restrict__ w2, const float* __restrict__ b2,
                                   const float* __restrict__ w3, const float* __restrict__ b3,
                                   const int* __restrict__ A, float* __restrict__ Ai) {
  int n = blockIdx.x * blockDim.x + threadIdx.x;
  if (n >= N_NODES) return;
  float em[16];
  int a = A[n];
#pragma unroll
  for (int i = 0; i < 16; ++i) em[i] = emb[a * 16 + i];
  float h1[64];
#pragma unroll
  for (int j = 0; j < 64; ++j) {
    float s = b1[j];
#pragma unroll
    for (int i = 0; i < 16; ++i) s = fmaf(em[i], w1[i * 64 + j], s);
    h1[j] = silu_f(s);
  }
  float h2[32];
#pragma unroll
  for (int j = 0; j < 32; ++j) {
    float s = b2[j];
#pragma unroll
    for (int i = 0; i < 64; ++i) s = fmaf(h1[i], w2[i * 32 + j], s);
    h2[j] = silu_f(s);
  }
#pragma unroll
  for (int j = 0; j < 8; ++j) {
    float s = b3[j];
#pragma unroll
    for (int i = 0; i < 32; ++i) s = fmaf(h2[i], w3[i * 8 + j], s);
    Ai[n * 8 + j] = s;
  }
}

// ------------------------------------------- one WMMA GEMM layer (K x 64), silu, transpose
// tb: 16 rows x 68 floats (stride pad for bank-conflict-free b64 access).
// A comes from tb (16 x K activations), B from packed weights, result (silu * oscale)
// stored transposed back into tb.
__device__ __forceinline__ void gemm_layer64(const float* __restrict__ Wp, int ksteps,
                                             float* tb, int lane, int hl, float oscale) {
  int m = lane & 15;
  v8f acc0 = {}, acc1 = {}, acc2 = {}, acc3 = {};
  const v2f* wp = (const v2f*)Wp;
  for (int ks = 0; ks < ksteps; ++ks) {
    int k0 = 4 * ks + 2 * hl;
    v2f a = *(const v2f*)(tb + m * 68 + k0);
    int base = ks * 4 * 32 + lane;
    acc0 = __builtin_amdgcn_wmma_f32_16x16x4_f32(false, a, false, wp[base +  0], (short)0, acc0, false, false);
    acc1 = __builtin_amdgcn_wmma_f32_16x16x4_f32(false, a, false, wp[base + 32], (short)0, acc1, false, false);
    acc2 = __builtin_amdgcn_wmma_f32_16x16x4_f32(false, a, false, wp[base + 64], (short)0, acc2, false, false);
    acc3 = __builtin_amdgcn_wmma_f32_16x16x4_f32(false, a, false, wp[base + 96], (short)0, acc3, false, false);
  }
#pragma unroll
  for (int j = 0; j < 8; ++j) {
    int er = j + 8 * hl;
    tb[er * 68 +  0 + m] = silu_f(acc0[j]) * oscale;
    tb[er * 68 + 16 + m] = silu_f(acc1[j]) * oscale;
    tb[er * 68 + 32 + m] = silu_f(acc2[j]) * oscale;
    tb[er * 68 + 48 + m] = silu_f(acc3[j]) * oscale;
  }
  lds_fence();
}

// ---------------------------------------------------------------- fused edge kernel
__global__ __launch_bounds__(128) void e3_edge_kernel(
    const float* __restrict__ pos,  const int* __restrict__ batch,
    const int* __restrict__ esrc,   const int* __restrict__ edst,
    const float* __restrict__ eshift, const float* __restrict__ cell,
    const float* __restrict__ Ai,
    const float* __restrict__ w1p,  const float* __restrict__ w2p,
    const float* __restrict__ w3p,  const float* __restrict__ w4p,
    float* __restrict__ out, float* __restrict__ cnt) {
  __shared__ float Tbuf_s[WPB][16 * 68];   // activations (padded stride)
  __shared__ float Pbuf_s[WPB][16 * 64];   // P outer products, later reused for S

  int wave = threadIdx.x >> 5;
  int lane = threadIdx.x & 31;
  int hl   = lane >> 4;
  int m    = lane & 15;
  int tile = blockIdx.x * WPB + wave;
  int e    = tile * 16 + m;
  float* tb = Tbuf_s[wave];
  float* pb = Pbuf_s[wave];

  // ---- geometry (both half-waves compute edge m; cheap, keeps sh in regs) ----
  int src = esrc[e], dst = edst[e];
  int bg  = batch[src];
  float sxv = eshift[e * 3 + 0], syv = eshift[e * 3 + 1], szv = eshift[e * 3 + 2];
  const float* cl = cell + bg * 9;
  float shx = sxv * cl[0] + syv * cl[3] + szv * cl[6];
  float shy = sxv * cl[1] + syv * cl[4] + szv * cl[7];
  float shz = sxv * cl[2] + syv * cl[5] + szv * cl[8];
  float evx = pos[dst * 3 + 0] - pos[src * 3 + 0] + shx;
  float evy = pos[dst * 3 + 1] - pos[src * 3 + 1] + shy;
  float evz = pos[dst * 3 + 2] - pos[src * 3 + 2] + shz;
  float r = sqrtf(evx * evx + evy * evy + evz * evz);
  float ri = 1.0f / fmaxf(r, 1e-9f);
  float x = evx * ri, y = evy * ri, z = evz * ri;
  const float s3 = 1.7320508075688772f, s15 = 3.872983346207417f, s5 = 2.23606797749979f;
  float sh1a = s3 * y, sh1b = s3 * z, sh1c = s3 * x;
  float sh2a = s15 * x * y, sh2b = s15 * y * z, sh2c = 0.5f * s5 * (3.0f * z * z - 1.0f);
  float sh2d = s15 * x * z, sh2e = 0.5f * s15 * (x * x - y * y);

  // ---- radial basis -> tb cols 0..15 ; includes /1.12*sqrt(16) and L1 1/sqrt(16) fold ----
  if (hl == 0) {
    const float step = 5.0f / 17.0f;
#pragma unroll
    for (int k = 0; k < 16; ++k) {
      float t = (r - (float)(k + 1) * step) / step;
      tb[m * 68 + k] = __expf(-t * t) * (1.0f / 1.12f);
    }
  }

  // ---- P = Ai[src] (u) outer Ai[dst] (v) into LDS: lanes<16 rows u=0..3, >=16 u=4..7 ----
  float ad[8];
#pragma unroll
  for (int v = 0; v < 8; ++v) ad[v] = Ai[dst * 8 + v];
#pragma unroll
  for (int uu = 0; uu < 4; ++uu) {
    float au = Ai[src * 8 + uu + 4 * hl];
    int u = uu + 4 * hl;
#pragma unroll
    for (int v = 0; v < 8; ++v) pb[m * 64 + u * 8 + v] = au * ad[v];
  }
  lds_fence();

  // ---- radial MLP: 16 -> 64 -> 64 (scale folds: 1/8 into each stored act, 1/64 on g3) ----
  gemm_layer64(w1p,  4, tb, lane, hl, 0.125f);       // g1/8
  gemm_layer64(w2p, 16, tb, lane, hl, 0.125f);       // g2/8
  gemm_layer64(w3p, 16, tb, lane, hl, 1.0f / 64.0f); // g3/64  (w/8 and S/8 folds)

  // ---- big layer (64 x 768) fused with P contraction ----
  // D column tile t: l = t/16, u = (t%16)/2, v = 4*(t&1) + (n>>2), wj = n&3
  v8f S0 = {}, S1 = {}, S2 = {};
  const v2f* w4 = (const v2f*)w4p;
#pragma unroll
  for (int l = 0; l < 3; ++l) {
    for (int tt = 0; tt < 16; ++tt) {
      int t = l * 16 + tt;
      v8f acc = {};
      for (int ks = 0; ks < 16; ++ks) {
        int k0 = 4 * ks + 2 * hl;
        v2f a = *(const v2f*)(tb + m * 68 + k0);
        acc = __builtin_amdgcn_wmma_f32_16x16x4_f32(false, a, false,
                  w4[(ks * 48 + t) * 32 + lane], (short)0, acc, false, false);
      }
      int u  = tt >> 1;
      int vv = (tt & 1) * 4 + (m >> 2);
#pragma unroll
      for (int j = 0; j < 8; ++j) {
        int er = j + 8 * hl;
        float p = pb[er * 64 + u * 8 + vv];
        if (l == 0) S0[j] += p * acc[j];
        else if (l == 1) S1[j] += p * acc[j];
        else S2[j] += p * acc[j];
      }
    }
  }

  // ---- reduce over v-groups (lanes n, n^4, n^8 share wj = n&3) ----
#pragma unroll
  for (int j = 0; j < 8; ++j) {
    S0[j] += __shfl_xor(S0[j], 4, 32); S0[j] += __shfl_xor(S0[j], 8, 32);
    S1[j] += __shfl_xor(S1[j], 4, 32); S1[j] += __shfl_xor(S1[j], 8, 32);
    S2[j] += __shfl_xor(S2[j], 4, 32); S2[j] += __shfl_xor(S2[j], 8, 32);
  }

  // ---- stage S[e][l*4+wj] into LDS (reuse pb) ----
  if (m < 4) {
    int wj = m;
#pragma unroll
    for (int j = 0; j < 8; ++j) {
      int er = j + 8 * hl;
      pb[er * 12 + 0 + wj] = S0[j];
      pb[er * 12 + 4 + wj] = S1[j];
      pb[er * 12 + 8 + wj] = S2[j];
    }
  }
  lds_fence();

  // ---- scatter ef into out[dst] (each lane: its edge m, half the 36 columns) ----
  float* orow = out + (long)dst * OUT_CH;
  if (hl == 0) {
#pragma unroll
    for (int c = 0; c < 4; ++c) atomicAdd(orow + c, pb[m * 12 + c]);        // out0
#pragma unroll
    for (int w = 0; w < 4; ++w) {                                           // out1
      float sv = pb[m * 12 + 4 + w];
      atomicAdd(orow + 4 + w * 3 + 0, sv * sh1a);
      atomicAdd(orow + 4 + w * 3 + 1, sv * sh1b);
      atomicAdd(orow + 4 + w * 3 + 2, sv * sh1c);
    }
    float sv2 = pb[m * 12 + 8 + 0];                                         // out2 w=0 i=0,1
    atomicAdd(orow + 16, sv2 * sh2a);
    atomicAdd(orow + 17, sv2 * sh2b);
    atomicAdd(cnt + dst, 1.0f);
  } else {
    float sv0 = pb[m * 12 + 8 + 0];                                         // out2 w=0 i=2..4
    atomicAdd(orow + 18, sv0 * sh2c);
    atomicAdd(orow + 19, sv0 * sh2d);
    atomicAdd(orow + 20, sv0 * sh2e);
#pragma unroll
    for (int w = 1; w < 4; ++w) {                                           // out2 w=1..3
      float sv = pb[m * 12 + 8 + w];
      atomicAdd(orow + 16 + w * 5 + 0, sv * sh2a);
      atomicAdd(orow + 16 + w * 5 + 1, sv * sh2b);
      atomicAdd(orow + 16 + w * 5 + 2, sv * sh2c);
      atomicAdd(orow + 16 + w * 5 + 3, sv * sh2d);
      atomicAdd(orow + 16 + w * 5 + 4, sv * sh2e);
    }
  }
}

// ---------------------------------------------------------------- segment mean
__global__ void e3_finalize_kernel(float* __restrict__ out, const float* __restrict__ cnt) {
  int i = blockIdx.x * blockDim.x + threadIdx.x;
  if (i >= N_NODES * OUT_CH) return;
  out[i] /= fmaxf(cnt[i / OUT_CH], 1.0f);
}

// ---------------------------------------------------------------- launch
extern "C" void kernel_launch(void* const* d_in, const int* in_sizes, int n_in,
                              void* d_out, int out_size, void* d_ws, size_t ws_size,
                              hipStream_t stream) {
  (void)in_sizes; (void)n_in; (void)out_size; (void)ws_size;
  const float* pos    = (const float*)d_in[0];
  const int*   A      = (const int*)d_in[1];
  const int*   batch  = (const int*)d_in[2];
  const int*   esrc   = (const int*)d_in[3];
  const int*   edst   = (const int*)d_in[4];
  const float* eshift = (const float*)d_in[5];
  const float* cell   = (const float*)d_in[6];
  const float* emb    = (const float*)d_in[7];
  const float* fw1    = (const float*)d_in[8];
  const float* fb1    = (const float*)d_in[9];
  const float* fw2    = (const float*)d_in[10];
  const float* fb2    = (const float*)d_in[11];
  const float* fw3    = (const float*)d_in[12];
  const float* fb3    = (const float*)d_in[13];
  const float* fcw1   = (const float*)d_in[14];
  const float* fcw2   = (const float*)d_in[15];
  const float* fcw3   = (const float*)d_in[16];
  const float* fcw4   = (const float*)d_in[17];
  float* out = (float*)d_out;
  char*  ws  = (char*)d_ws;

  float* Ai  = (float*)(ws + 0);        // 20000*8*4  = 640000
  float* cnt = (float*)(ws + 640000);   // 20000*4    =  80000
  float* w1p = (float*)(ws + 720000);   // 16*64*4    =   4096
  float* w2p = (float*)(ws + 724096);   // 64*64*4    =  16384
  float* w3p = (float*)(ws + 740480);   // 64*64*4    =  16384
  float* w4p = (float*)(ws + 756864);   // 64*768*4   = 196608

  e3_zero_kernel<<<(N_NODES * OUT_CH + 255) / 256, 256, 0, stream>>>(out, cnt);
  e3_pack_kernel<<<(16 * 64 + 255) / 256, 256, 0, stream>>>(fcw1, w1p, 16, 64);
  e3_pack_kernel<<<(64 * 64 + 255) / 256, 256, 0, stream>>>(fcw2, w2p, 64, 64);
  e3_pack_kernel<<<(64 * 64 + 255) / 256, 256, 0, stream>>>(fcw3, w3p, 64, 64);
  e3_pack_kernel<<<(64 * 768 + 255) / 256, 256, 0, stream>>>(fcw4, w4p, 64, 768);
  e3_node_mlp_kernel<<<(N_NODES + 127) / 128, 128, 0, stream>>>(emb, fw1, fb1, fw2, fb2,
                                                                fw3, fb3, A, Ai);
  e3_edge_kernel<<<N_EDGES / (16 * WPB), 128, 0, stream>>>(pos, batch, esrc, edst, eshift,
                                                           cell, Ai, w1p, w2p, w3p, w4p,
                                                           out, cnt);
  e3_finalize_kernel<<<(N_NODES * OUT_CH + 255) / 256, 256, 0, stream>>>(out, cnt);
}